// DeepseekV2LiteMatAbsorbDecode_59992103190714
// MI455X (gfx1250) — compile-verified
//
#include <hip/hip_runtime.h>

// ---------------- problem constants ----------------
#define H_     2048
#define NH_    16
#define DR_    64
#define DC_    512
#define DV_    128
#define DN_    128
#define DQ_    192
#define BSZ_   32
#define KVLEN_ 4096
#define NQ_    (NH_ * DQ_)     // 3072
#define DKVA_  (DC_ + DR_)     // 576
#define SCALE_ 0.07216878364870323f   // 192^-0.5
#define EPS_   1e-6f

#define NCHUNK_ 8
#define CHUNK_  512
#define PANEL_  64
#define QS_     580            // padded LDS row stride (floats); 2320B = 16B-aligned

typedef float v2f __attribute__((ext_vector_type(2)));
typedef float v8f __attribute__((ext_vector_type(8)));

__device__ __forceinline__ v8f v8zero() {
  v8f z = {0.f,0.f,0.f,0.f,0.f,0.f,0.f,0.f};
  return z;
}

// D(16x16,f32) = A(16x4,f32) * B(4x16,f32) + C  — CDNA5 f32 tensor op
__device__ __forceinline__ v8f wmma4(v2f a, v2f b, v8f c) {
  return __builtin_amdgcn_wmma_f32_16x16x4_f32(false, a, false, b, (short)0, c,
                                               false, false);
}

// CDNA5 async global->LDS copy (ASYNCcnt tracked). Per-lane LDS destination
// address comes from a VGPR: low 32 bits of the flat shared-memory pointer are
// the LDS offset (flat LDS aperture truncation, per ISA 10.2).
__device__ __forceinline__ void async_copy16(const float* __restrict__ src,
                                             float* dst_lds) {
  asm volatile("global_load_async_to_lds_b128 %0, %1, off"
               :
               : "v"((unsigned)(uintptr_t)dst_lds), "v"(src)
               : "memory");
}
__device__ __forceinline__ void wait_async0() {
  asm volatile("s_wait_asynccnt 0x0" ::: "memory");
}

// ---------------------------------------------------------------------------
// q_full(32x3072) = hidden(32x2048) @ W_UQR(2048x3072)
// one wave per 16x16 output tile; 2 M-tiles x 192 N-tiles = 384 waves
// ---------------------------------------------------------------------------
__global__ __launch_bounds__(128) void k_qproj(const float* __restrict__ hs,
                                               const float* __restrict__ w,
                                               float* __restrict__ q_full) {
  const int tid = threadIdx.x, wave = tid >> 5, lane = tid & 31;
  const int half = lane >> 4, l16 = lane & 15;
  const int gw = blockIdx.x * 4 + wave;       // 0..383
  const int mt = gw / 192, nt = gw % 192;
  const int m0 = mt * 16, n0 = nt * 16;
  v8f acc = v8zero();
  const float* arow = hs + (size_t)(m0 + l16) * H_ + 2 * half;
  const float* bcol = w + (size_t)(2 * half) * NQ_ + n0 + l16;
#pragma unroll 4
  for (int kb = 0; kb < H_; kb += 4) {
    v2f a = *(const v2f*)(arow + kb);
    v2f b; b.x = bcol[(size_t)kb * NQ_]; b.y = bcol[(size_t)(kb + 1) * NQ_];
    acc = wmma4(a, b, acc);
  }
  float* drow = q_full + n0 + l16;
#pragma unroll
  for (int r = 0; r < 8; ++r)
    drow[(size_t)(m0 + r + 8 * half) * NQ_] = acc[r];
}

// ---------------------------------------------------------------------------
// latent(32x576) = hidden(32x2048) @ W_kva(2048x576); 2 x 36 = 72 waves
// ---------------------------------------------------------------------------
__global__ __launch_bounds__(128) void k_latent(const float* __restrict__ hs,
                                                const float* __restrict__ w,
                                                float* __restrict__ latent) {
  const int tid = threadIdx.x, wave = tid >> 5, lane = tid & 31;
  const int half = lane >> 4, l16 = lane & 15;
  const int gw = blockIdx.x * 4 + wave;       // 0..71
  const int mt = gw / 36, nt = gw % 36;
  const int m0 = mt * 16, n0 = nt * 16;
  v8f acc = v8zero();
  const float* arow = hs + (size_t)(m0 + l16) * H_ + 2 * half;
  const float* bcol = w + (size_t)(2 * half) * DKVA_ + n0 + l16;
#pragma unroll 4
  for (int kb = 0; kb < H_; kb += 4) {
    v2f a = *(const v2f*)(arow + kb);
    v2f b; b.x = bcol[(size_t)kb * DKVA_]; b.y = bcol[(size_t)(kb + 1) * DKVA_];
    acc = wmma4(a, b, acc);
  }
  float* drow = latent + n0 + l16;
#pragma unroll
  for (int r = 0; r < 8; ++r)
    drow[(size_t)(m0 + r + 8 * half) * DKVA_] = acc[r];
}

// ---------------------------------------------------------------------------
// RoPE on q_pe: q_pe_ws[b,h,r] from q_full[b, h*192+128+r]
// ---------------------------------------------------------------------------
__global__ __launch_bounds__(256) void k_rope_q(const float* __restrict__ q_full,
                                                const float* __restrict__ cosv,
                                                const float* __restrict__ sinv,
                                                float* __restrict__ q_pe) {
  int idx = blockIdx.x * 256 + threadIdx.x;           // 32*16*64
  if (idx >= BSZ_ * NH_ * DR_) return;
  int b = idx / (NH_ * DR_);
  int h = (idx / DR_) % NH_;
  int r = idx % DR_;
  const float* qp = q_full + (size_t)b * NQ_ + h * DQ_ + DN_;
  float x = qp[r];
  float rot = (r < DR_ / 2) ? -qp[r + DR_ / 2] : qp[r - DR_ / 2];
  q_pe[idx] = x * cosv[r] + rot * sinv[r];
}

// ---------------------------------------------------------------------------
// RMS-norm of latent[:, :512] -> c_new ; RoPE of latent[:, 512:] -> kpe_new
// ---------------------------------------------------------------------------
__global__ __launch_bounds__(256) void k_normrope(const float* __restrict__ latent,
                                                  const float* __restrict__ ln_w,
                                                  const float* __restrict__ cosv,
                                                  const float* __restrict__ sinv,
                                                  float* __restrict__ c_new,
                                                  float* __restrict__ kpe_new) {
  __shared__ float red[256];
  __shared__ float srs;
  const int b = blockIdx.x, tid = threadIdx.x;
  const float* lat = latent + (size_t)b * DKVA_;
  float a0 = lat[tid], a1 = lat[tid + 256];
  red[tid] = a0 * a0 + a1 * a1;
  __syncthreads();
  for (int off = 128; off > 0; off >>= 1) {
    if (tid < off) red[tid] += red[tid + off];
    __syncthreads();
  }
  if (tid == 0) srs = rsqrtf(red[0] / (float)DC_ + EPS_);
  __syncthreads();
  float rs = srs;
  c_new[(size_t)b * DC_ + tid]       = ln_w[tid] * a0 * rs;
  c_new[(size_t)b * DC_ + tid + 256] = ln_w[tid + 256] * a1 * rs;
  if (tid < DR_) {
    float x = lat[DC_ + tid];
    float rot = (tid < DR_ / 2) ? -lat[DC_ + tid + DR_ / 2]
                                : lat[DC_ + tid - DR_ / 2];
    kpe_new[(size_t)b * DR_ + tid] = x * cosv[tid] + rot * sinv[tid];
  }
}

// ---------------------------------------------------------------------------
// absorption: q_abs[b,h,c] = sum_d q_nope[b,h,d] * W_UK[h,c,d]
// per head M=32,N=512,K=128 -> 16h * 2mt * 32nt = 1024 waves
// ---------------------------------------------------------------------------
__global__ __launch_bounds__(128) void k_absorb(const float* __restrict__ q_full,
                                                const float* __restrict__ w_uk,
                                                float* __restrict__ q_abs) {
  const int tid = threadIdx.x, wave = tid >> 5, lane = tid & 31;
  const int half = lane >> 4, l16 = lane & 15;
  const int gw = blockIdx.x * 4 + wave;       // 0..1023
  const int h = gw / 64, rem = gw % 64;
  const int mt = rem / 32, nt = rem % 32;
  const int m0 = mt * 16, n0 = nt * 16;
  v8f acc = v8zero();
  const float* arow = q_full + (size_t)(m0 + l16) * NQ_ + h * DQ_ + 2 * half;
  const float* brow = w_uk + ((size_t)h * DC_ + n0 + l16) * DN_ + 2 * half;
#pragma unroll 8
  for (int kb = 0; kb < DN_; kb += 4) {
    v2f a = *(const v2f*)(arow + kb);
    v2f b = *(const v2f*)(brow + kb);      // W_UK[h, n0+l16, kb..kb+1] contiguous
    acc = wmma4(a, b, acc);
  }
#pragma unroll
  for (int r = 0; r < 8; ++r)
    q_abs[((size_t)(m0 + r + 8 * half) * NH_ + h) * DC_ + n0 + l16] = acc[r];
}

// ---------------------------------------------------------------------------
// flash-decoding attention partials. grid (b=32, chunk=8), 128 threads.
// Q(16x576) vs KV rows; panels of 64 rows async-staged into LDS; online softmax.
// ---------------------------------------------------------------------------
__global__ __launch_bounds__(128) void k_attn(const float* __restrict__ ckv,
                                              const float* __restrict__ kpe,
                                              const float* __restrict__ q_abs,
                                              const float* __restrict__ q_pe,
                                              const float* __restrict__ c_new,
                                              const float* __restrict__ kpe_new,
                                              float* __restrict__ part_m,
                                              float* __restrict__ part_l,
                                              float* __restrict__ part_acc) {
  __shared__ float lds_q[16 * QS_];
  __shared__ float lds_kv[PANEL_ * QS_];
  __shared__ float lds_p[16 * 68];
  __shared__ float lds_m[16], lds_l[16], lds_fac[16];

  const int b = blockIdx.x, ch = blockIdx.y;
  const int tid = threadIdx.x, wave = tid >> 5, lane = tid & 31;
  const int half = lane >> 4, l16 = lane & 15;

  // async-stage combined Q = [q_abs | q_pe] (16 heads x 576), 16B granules
  for (int i = tid; i < 16 * 144; i += 128) {
    int h = i / 144, q4 = i % 144;
    const float* src = (q4 < 128)
        ? q_abs + ((size_t)b * NH_ + h) * DC_ + q4 * 4
        : q_pe + ((size_t)b * NH_ + h) * DR_ + (q4 - 128) * 4;
    async_copy16(src, lds_q + h * QS_ + q4 * 4);
  }
  if (tid < 16) { lds_m[tid] = -3.0e38f; lds_l[tid] = 0.f; }

  v8f acc[8];
#pragma unroll
  for (int t = 0; t < 8; ++t) acc[t] = v8zero();

  const int k_base = ch * CHUNK_;
  for (int p = 0; p < CHUNK_ / PANEL_; ++p) {
    const int r0 = k_base + p * PANEL_;
    __syncthreads();   // previous panel fully consumed (also covers lds_m/l init)

    // async-stage 64 rows x (512 ckv + 64 kpe) into LDS, 16B granules
    for (int i = tid; i < PANEL_ * 144; i += 128) {
      int row = i / 144, q4 = i % 144;
      const float* src =
          (q4 < 128)
              ? (ckv + ((size_t)b * KVLEN_ + r0 + row) * DC_ + q4 * 4)
              : (kpe + ((size_t)b * KVLEN_ + r0 + row) * DR_ + (q4 - 128) * 4);
      async_copy16(src, lds_kv + row * QS_ + q4 * 4);
    }
    // speculative stream prefetch of next panel (global_prefetch_b8)
    if (p < CHUNK_ / PANEL_ - 1)
      __builtin_prefetch(ckv + ((size_t)b * KVLEN_ + r0 + PANEL_ + (tid >> 1)) * DC_ +
                             (tid & 1) * 256, 0, 1);
    wait_async0();       // wave's async copies (incl. Q on first panel) complete
    __syncthreads();     // all waves' staging visible

    // reference overwrites cache row KVLEN-1 with fresh c / k_pe
    if (r0 + PANEL_ == KVLEN_) {
      for (int i = tid; i < DKVA_; i += 128) {
        float v = (i < DC_) ? c_new[(size_t)b * DC_ + i]
                            : kpe_new[(size_t)b * DR_ + (i - DC_)];
        lds_kv[(PANEL_ - 1) * QS_ + i] = v;
      }
      __syncthreads();
    }

    // scores: wave w computes 16(heads) x 16(kv) tile at n0 = w*16
    {
      const int n0 = wave * 16;
      v8f s = v8zero();
      const float* qrow = lds_q + l16 * QS_ + 2 * half;
      const float* krow = lds_kv + (n0 + l16) * QS_ + 2 * half;
#pragma unroll 8
      for (int kb = 0; kb < DKVA_; kb += 4) {
        v2f a = *(const v2f*)(qrow + kb);
        v2f bb = *(const v2f*)(krow + kb);
        s = wmma4(a, bb, s);
      }
#pragma unroll
      for (int r = 0; r < 8; ++r)
        lds_p[(r + 8 * half) * 68 + n0 + l16] = s[r] * SCALE_;
    }
    __syncthreads();

    // online softmax update (one lane per head)
    if (wave == 0 && lane < 16) {
      const int h = lane;
      float mold = lds_m[h], mx = mold;
      for (int j = 0; j < PANEL_; ++j) mx = fmaxf(mx, lds_p[h * 68 + j]);
      float fac = __expf(mold - mx);
      float sum = 0.f;
      for (int j = 0; j < PANEL_; ++j) {
        float pe = __expf(lds_p[h * 68 + j] - mx);
        lds_p[h * 68 + j] = pe;
        sum += pe;
      }
      lds_l[h] = lds_l[h] * fac + sum;
      lds_m[h] = mx;
      lds_fac[h] = fac;
    }
    __syncthreads();

    // acc rescale + P(16x64) x V(64x512); wave owns 8 N-tiles of 16
#pragma unroll
    for (int t = 0; t < 8; ++t) {
      const int c0 = (wave * 8 + t) * 16;
      v8f a8 = acc[t];
#pragma unroll
      for (int r = 0; r < 8; ++r) a8[r] *= lds_fac[r + 8 * half];
      const float* prow = lds_p + l16 * 68 + 2 * half;
#pragma unroll
      for (int kb = 0; kb < PANEL_; kb += 4) {
        v2f a = *(const v2f*)(prow + kb);
        v2f bb;
        bb.x = lds_kv[(kb + 2 * half) * QS_ + c0 + l16];
        bb.y = lds_kv[(kb + 2 * half + 1) * QS_ + c0 + l16];
        a8 = wmma4(a, bb, a8);
      }
      acc[t] = a8;
    }
  }
  __syncthreads();

  const int pi = (b * NCHUNK_ + ch) * NH_;
  if (wave == 0 && lane < 16) {
    part_m[pi + lane] = lds_m[lane];
    part_l[pi + lane] = lds_l[lane];
  }
#pragma unroll
  for (int t = 0; t < 8; ++t) {
    const int c0 = (wave * 8 + t) * 16;
#pragma unroll
    for (int r = 0; r < 8; ++r)
      part_acc[(size_t)(pi + r + 8 * half) * DC_ + c0 + l16] = acc[t][r];
  }
}

// ---------------------------------------------------------------------------
// combine chunk partials -> attn[b,h,c]
// ---------------------------------------------------------------------------
__global__ __launch_bounds__(128) void k_reduce(const float* __restrict__ part_m,
                                                const float* __restrict__ part_l,
                                                const float* __restrict__ part_acc,
                                                float* __restrict__ attn) {
  const int b = blockIdx.x, h = blockIdx.y, tid = threadIdx.x;
  float M = -3.0e38f;
  for (int j = 0; j < NCHUNK_; ++j)
    M = fmaxf(M, part_m[(b * NCHUNK_ + j) * NH_ + h]);
  float wgt[NCHUNK_];
  float denom = 0.f;
  for (int j = 0; j < NCHUNK_; ++j) {
    wgt[j] = __expf(part_m[(b * NCHUNK_ + j) * NH_ + h] - M);
    denom += wgt[j] * part_l[(b * NCHUNK_ + j) * NH_ + h];
  }
  float inv = 1.0f / denom;
  for (int c = tid; c < DC_; c += 128) {
    float s = 0.f;
    for (int j = 0; j < NCHUNK_; ++j)
      s += wgt[j] * part_acc[(size_t)((b * NCHUNK_ + j) * NH_ + h) * DC_ + c];
    attn[((size_t)b * NH_ + h) * DC_ + c] = s * inv;
  }
}

// ---------------------------------------------------------------------------
// out_uv[b, h*128+v] = sum_c attn[b,h,c] * W_UV[h,c,v]; 16h*2mt*8nt = 256 waves
// ---------------------------------------------------------------------------
__global__ __launch_bounds__(128) void k_uv(const float* __restrict__ attn,
                                            const float* __restrict__ w_uv,
                                            float* __restrict__ out_uv) {
  const int tid = threadIdx.x, wave = tid >> 5, lane = tid & 31;
  const int half = lane >> 4, l16 = lane & 15;
  const int gw = blockIdx.x * 4 + wave;       // 0..255
  const int h = gw / 16, rem = gw % 16;
  const int mt = rem / 8, nt = rem % 8;
  const int m0 = mt * 16, n0 = nt * 16;
  v8f acc = v8zero();
  const float* arow = attn + ((size_t)(m0 + l16) * NH_ + h) * DC_ + 2 * half;
  const float* bcol = w_uv + ((size_t)h * DC_ + 2 * half) * DV_ + n0 + l16;
#pragma unroll 8
  for (int kb = 0; kb < DC_; kb += 4) {
    v2f a = *(const v2f*)(arow + kb);
    v2f b; b.x = bcol[(size_t)kb * DV_]; b.y = bcol[(size_t)(kb + 1) * DV_];
    acc = wmma4(a, b, acc);
  }
#pragma unroll
  for (int r = 0; r < 8; ++r)
    out_uv[(size_t)(m0 + r + 8 * half) * (NH_ * DV_) + h * DV_ + n0 + l16] = acc[r];
}

// ---------------------------------------------------------------------------
// final: out(32x2048) = out_uv(32x2048) @ W_O(2048x2048); 2mt*128nt = 256 waves
// ---------------------------------------------------------------------------
__global__ __launch_bounds__(128) void k_out(const float* __restrict__ out_uv,
                                             const float* __restrict__ w_o,
                                             float* __restrict__ out) {
  const int tid = threadIdx.x, wave = tid >> 5, lane = tid & 31;
  const int half = lane >> 4, l16 = lane & 15;
  const int gw = blockIdx.x * 4 + wave;       // 0..255
  const int mt = gw / 128, nt = gw % 128;
  const int m0 = mt * 16, n0 = nt * 16;
  v8f acc = v8zero();
  const float* arow = out_uv + (size_t)(m0 + l16) * H_ + 2 * half;
  const float* bcol = w_o + (size_t)(2 * half) * H_ + n0 + l16;
#pragma unroll 4
  for (int kb = 0; kb < (NH_ * DV_); kb += 4) {
    v2f a = *(const v2f*)(arow + kb);
    v2f b; b.x = bcol[(size_t)kb * H_]; b.y = bcol[(size_t)(kb + 1) * H_];
    acc = wmma4(a, b, acc);
  }
#pragma unroll
  for (int r = 0; r < 8; ++r)
    out[(size_t)(m0 + r + 8 * half) * H_ + n0 + l16] = acc[r];
}

// ---------------------------------------------------------------------------
extern "C" void kernel_launch(void* const* d_in, const int* in_sizes, int n_in,
                              void* d_out, int out_size, void* d_ws, size_t ws_size,
                              hipStream_t stream) {
  const float* hs    = (const float*)d_in[0];   // (32,2048)
  const float* ckv   = (const float*)d_in[1];   // (32,4096,512)
  const float* kpe   = (const float*)d_in[2];   // (32,4096,64)
  const float* w_uqr = (const float*)d_in[3];   // (2048,3072)
  const float* w_kva = (const float*)d_in[4];   // (2048,576)
  const float* ln_w  = (const float*)d_in[5];   // (512,)
  const float* w_uk  = (const float*)d_in[6];   // (16,512,128)
  const float* w_uv  = (const float*)d_in[7];   // (16,512,128)
  const float* w_o   = (const float*)d_in[8];   // (2048,2048)
  const float* cosv  = (const float*)d_in[9];   // (1,64)
  const float* sinv  = (const float*)d_in[10];  // (1,64)
  float* out = (float*)d_out;                   // (32,2048)

  float* ws = (float*)d_ws;
  float* q_full   = ws;                         // 32*3072
  float* q_pe     = q_full + BSZ_ * NQ_;        // 32*16*64
  float* latent   = q_pe + BSZ_ * NH_ * DR_;    // 32*576
  float* c_new    = latent + BSZ_ * DKVA_;      // 32*512
  float* kpe_new  = c_new + BSZ_ * DC_;         // 32*64
  float* q_abs    = kpe_new + BSZ_ * DR_;       // 32*16*512
  float* part_m   = q_abs + (size_t)BSZ_ * NH_ * DC_;       // 32*8*16
  float* part_l   = part_m + BSZ_ * NCHUNK_ * NH_;
  float* part_acc = part_l + BSZ_ * NCHUNK_ * NH_;          // 32*8*16*512
  float* attn     = part_acc + (size_t)BSZ_ * NCHUNK_ * NH_ * DC_;  // 32*16*512
  float* out_uv   = attn + (size_t)BSZ_ * NH_ * DC_;        // 32*2048

  k_qproj<<<96, 128, 0, stream>>>(hs, w_uqr, q_full);
  k_latent<<<18, 128, 0, stream>>>(hs, w_kva, latent);
  k_rope_q<<<(BSZ_ * NH_ * DR_ + 255) / 256, 256, 0, stream>>>(q_full, cosv, sinv, q_pe);
  k_normrope<<<BSZ_, 256, 0, stream>>>(latent, ln_w, cosv, sinv, c_new, kpe_new);
  k_absorb<<<256, 128, 0, stream>>>(q_full, w_uk, q_abs);
  k_attn<<<dim3(BSZ_, NCHUNK_), 128, 0, stream>>>(ckv, kpe, q_abs, q_pe, c_new,
                                                  kpe_new, part_m, part_l, part_acc);
  k_reduce<<<dim3(BSZ_, NH_), 128, 0, stream>>>(part_m, part_l, part_acc, attn);
  k_uv<<<64, 128, 0, stream>>>(attn, w_uv, out_uv);
  k_out<<<64, 128, 0, stream>>>(out_uv, w_o, out);
}